// get_model_65309272703162
// MI455X (gfx1250) — compile-verified
//
#include <hip/hip_runtime.h>
#include <hip/hip_bf16.h>
#include <math.h>

typedef float v2f __attribute__((ext_vector_type(2)));
typedef float v8f __attribute__((ext_vector_type(8)));

#define BATCH 8
#define NPTS  4096

// ---------------------------------------------------------------------------
// input transpose: xyz (B,9,N) -> pts0 (B,N,9), xyz0 (B,N,3)
// ---------------------------------------------------------------------------
__global__ void k_transpose_in(const float* __restrict__ x,
                               float* __restrict__ pts0,
                               float* __restrict__ xyz0) {
  int idx = blockIdx.x * blockDim.x + threadIdx.x;
  if (idx >= BATCH * NPTS) return;
  int b = idx / NPTS, n = idx % NPTS;
  const float* src = x + (size_t)b * 9 * NPTS + n;
  float* pd = pts0 + (size_t)idx * 9;
  float* xd = xyz0 + (size_t)idx * 3;
#pragma unroll
  for (int c = 0; c < 9; ++c) pd[c] = src[(size_t)c * NPTS];
  xd[0] = pd[0]; xd[1] = pd[1]; xd[2] = pd[2];
}

// ---------------------------------------------------------------------------
// farthest point sampling: one block per batch, serial argmax rounds
// ---------------------------------------------------------------------------
__global__ void k_fps(const float* __restrict__ xyz, int N, int P,
                      int* __restrict__ fidx) {
  __shared__ float dist[4096];
  __shared__ float rv[256];
  __shared__ int   ri[256];
  __shared__ float cx, cy, cz;
  int b = blockIdx.x;
  int t = threadIdx.x;
  const float* xb = xyz + (size_t)b * N * 3;
  for (int i = t; i < N; i += blockDim.x) dist[i] = 1e10f;
  __syncthreads();
  int far = 0;
  for (int s = 0; s < P; ++s) {
    if (t == 0) {
      fidx[b * P + s] = far;
      cx = xb[far * 3 + 0]; cy = xb[far * 3 + 1]; cz = xb[far * 3 + 2];
    }
    __syncthreads();
    float bv = -1.0f; int bi = 0;
    for (int i = t; i < N; i += blockDim.x) {
      float dx = xb[i * 3 + 0] - cx;
      float dy = xb[i * 3 + 1] - cy;
      float dz = xb[i * 3 + 2] - cz;
      float d = dx * dx + dy * dy + dz * dz;
      float dm = dist[i];
      if (d < dm) { dm = d; dist[i] = dm; }
      if (dm > bv) { bv = dm; bi = i; }
    }
    rv[t] = bv; ri[t] = bi;
    __syncthreads();
    for (int k = 128; k > 0; k >>= 1) {
      if (t < k) {
        if (rv[t + k] > rv[t] || (rv[t + k] == rv[t] && ri[t + k] < ri[t])) {
          rv[t] = rv[t + k]; ri[t] = ri[t + k];
        }
      }
      __syncthreads();
    }
    far = ri[0];
    __syncthreads();
  }
}

// gather sampled coordinates: lx[b,p,:] = xyz[b, fidx[b,p], :]
__global__ void k_gather3(const float* __restrict__ xyz, int N,
                          const int* __restrict__ fidx, int P,
                          float* __restrict__ lx) {
  int idx = blockIdx.x * blockDim.x + threadIdx.x;
  if (idx >= BATCH * P) return;
  int b = idx / P;
  int j = fidx[idx];
  const float* s = xyz + ((size_t)b * N + j) * 3;
  float* d = lx + (size_t)idx * 3;
  d[0] = s[0]; d[1] = s[1]; d[2] = s[2];
}

// ---------------------------------------------------------------------------
// ball query: first S indices with d^2 <= r^2, pad with first hit
// ---------------------------------------------------------------------------
__global__ void k_ballquery(const float* __restrict__ xyz,
                            const float* __restrict__ new_xyz, float r2,
                            int N, int P, int S, int* __restrict__ gidx) {
  int idx = blockIdx.x * blockDim.x + threadIdx.x;
  if (idx >= BATCH * P) return;
  int b = idx / P;
  const float* xb = xyz + (size_t)b * N * 3;
  float qx = new_xyz[idx * 3 + 0];
  float qy = new_xyz[idx * 3 + 1];
  float qz = new_xyz[idx * 3 + 2];
  int* out = gidx + (size_t)idx * S;
  int cnt = 0, first = 0;
  for (int j = 0; j < N && cnt < S; ++j) {
    float dx = xb[j * 3 + 0] - qx;
    float dy = xb[j * 3 + 1] - qy;
    float dz = xb[j * 3 + 2] - qz;
    float d = dx * dx + dy * dy + dz * dz;
    if (d <= r2) {
      if (cnt == 0) first = j;
      out[cnt++] = j;
    }
  }
  for (; cnt < S; ++cnt) out[cnt] = first;
}

// grouping: rows (B*P*S) x (3 + Cpts) = [xyz - new_xyz, pts[gidx]]
__global__ void k_group(const float* __restrict__ xyz,
                        const float* __restrict__ pts, int Cpts, int N,
                        const float* __restrict__ new_xyz,
                        const int* __restrict__ gidx, int P, int S,
                        float* __restrict__ out) {
  int idx = blockIdx.x * blockDim.x + threadIdx.x;
  int total = BATCH * P * S;
  if (idx >= total) return;
  int s  = idx % S;
  int bp = idx / S;
  int b  = bp / P;
  int j  = gidx[(size_t)bp * S + s];
  int K  = 3 + Cpts;
  float* row = out + (size_t)idx * K;
  const float* xb = xyz + ((size_t)b * N + j) * 3;
  row[0] = xb[0] - new_xyz[bp * 3 + 0];
  row[1] = xb[1] - new_xyz[bp * 3 + 1];
  row[2] = xb[2] - new_xyz[bp * 3 + 2];
  const float* pb = pts + ((size_t)b * N + j) * Cpts;
  for (int c = 0; c < Cpts; ++c) row[3 + c] = pb[c];
}

// ---------------------------------------------------------------------------
// WMMA f32 GEMM: Y[r,o] = sum_k X[r,k] * W[o,k] + bias[o]
// One wave computes a 32x32 output tile as 2x2 WMMA_F32_16X16X4_F32 tiles
// (4 accumulators, A/B fragments each reused twice). Block = 4 waves = 64x64.
// Main K-loop has NO guards (all loads in-bounds); K%4 tail uses clamped
// indices + value-select so no exec-mask branching appears in the hot path.
// All R in this network are multiples of 32, so rows need no masking.
// ---------------------------------------------------------------------------
__global__ void k_gemm_bias(const float* __restrict__ X,
                            const float* __restrict__ W,
                            const float* __restrict__ bias,
                            float* __restrict__ Y,
                            int R, int K, int O) {
  int lane = threadIdx.x & 31;
  int wave = threadIdx.x >> 5;
  int wr = wave & 1, wc = wave >> 1;
  int row0 = blockIdx.x * 64 + wr * 32;
  int col0 = blockIdx.y * 64 + wc * 32;
  int Rpad = (R + 31) & ~31;
  int Opad = (O + 31) & ~31;
  if (row0 >= Rpad || col0 >= Opad) return;   // wave-uniform: EXEC stays all-1

  int m     = lane & 15;
  int khalf = lane >> 4;
  int kbase = 2 * khalf;

  // clamped source rows (always valid addresses; OOB cols are never stored)
  int rA0 = row0 + m;       if (rA0 > R - 1) rA0 = R - 1;
  int rA1 = row0 + 16 + m;  if (rA1 > R - 1) rA1 = R - 1;
  int cB0 = col0 + m;       if (cB0 > O - 1) cB0 = O - 1;
  int cB1 = col0 + 16 + m;  if (cB1 > O - 1) cB1 = O - 1;
  const float* Xr0 = X + (size_t)rA0 * K;
  const float* Xr1 = X + (size_t)rA1 * K;
  const float* Wc0 = W + (size_t)cB0 * K;
  const float* Wc1 = W + (size_t)cB1 * K;

  v8f acc00 = {0.f,0.f,0.f,0.f,0.f,0.f,0.f,0.f};
  v8f acc01 = acc00, acc10 = acc00, acc11 = acc00;

  int K4 = K & ~3;
#pragma unroll 4
  for (int k0 = 0; k0 < K4; k0 += 4) {
    int ka = k0 + kbase;
    v2f a0, a1, b0, b1;
    a0[0] = Xr0[ka]; a0[1] = Xr0[ka + 1];
    a1[0] = Xr1[ka]; a1[1] = Xr1[ka + 1];
    b0[0] = Wc0[ka]; b0[1] = Wc0[ka + 1];
    b1[0] = Wc1[ka]; b1[1] = Wc1[ka + 1];
    acc00 = __builtin_amdgcn_wmma_f32_16x16x4_f32(false, a0, false, b0, (short)0, acc00, false, false);
    acc01 = __builtin_amdgcn_wmma_f32_16x16x4_f32(false, a0, false, b1, (short)0, acc01, false, false);
    acc10 = __builtin_amdgcn_wmma_f32_16x16x4_f32(false, a1, false, b0, (short)0, acc10, false, false);
    acc11 = __builtin_amdgcn_wmma_f32_16x16x4_f32(false, a1, false, b1, (short)0, acc11, false, false);
  }
  if (K4 < K) {
    int ka = K4 + kbase;
    int k0c = (ka     < K) ? ka     : K - 1;
    int k1c = (ka + 1 < K) ? ka + 1 : K - 1;
    float m0 = (ka     < K) ? 1.f : 0.f;
    float m1 = (ka + 1 < K) ? 1.f : 0.f;
    float x00 = Xr0[k0c], x01 = Xr0[k1c];
    float x10 = Xr1[k0c], x11 = Xr1[k1c];
    float w00 = Wc0[k0c], w01 = Wc0[k1c];
    float w10 = Wc1[k0c], w11 = Wc1[k1c];
    v2f a0, a1, b0, b1;
    a0[0] = x00 * m0; a0[1] = x01 * m1;
    a1[0] = x10 * m0; a1[1] = x11 * m1;
    b0[0] = w00 * m0; b0[1] = w01 * m1;
    b1[0] = w10 * m0; b1[1] = w11 * m1;
    acc00 = __builtin_amdgcn_wmma_f32_16x16x4_f32(false, a0, false, b0, (short)0, acc00, false, false);
    acc01 = __builtin_amdgcn_wmma_f32_16x16x4_f32(false, a0, false, b1, (short)0, acc01, false, false);
    acc10 = __builtin_amdgcn_wmma_f32_16x16x4_f32(false, a1, false, b0, (short)0, acc10, false, false);
    acc11 = __builtin_amdgcn_wmma_f32_16x16x4_f32(false, a1, false, b1, (short)0, acc11, false, false);
  }

  // epilogue: D layout -> Y[row0 + 16*i + v + 8*khalf][col0 + 16*j + m]
  int nD0 = col0 + m;
  int nD1 = col0 + 16 + m;
  if (nD0 < O) {
    float bv = bias ? bias[nD0] : 0.f;
#pragma unroll
    for (int v = 0; v < 8; ++v) {
      int mD = v + 8 * khalf;
      Y[(size_t)(row0 + mD) * O + nD0]      = acc00[v] + bv;
      Y[(size_t)(row0 + 16 + mD) * O + nD0] = acc10[v] + bv;
    }
  }
  if (nD1 < O) {
    float bv = bias ? bias[nD1] : 0.f;
#pragma unroll
    for (int v = 0; v < 8; ++v) {
      int mD = v + 8 * khalf;
      Y[(size_t)(row0 + mD) * O + nD1]      = acc01[v] + bv;
      Y[(size_t)(row0 + 16 + mD) * O + nD1] = acc11[v] + bv;
    }
  }
}

// ---------------------------------------------------------------------------
// batch-norm: per-channel stats over all rows (deterministic block reduce)
// ---------------------------------------------------------------------------
__global__ void k_bn_stats(const float* __restrict__ Y, int R, int O,
                           float* __restrict__ stats) {
  __shared__ float s1[256], s2[256];
  int c = blockIdx.x;
  float sum = 0.f, sq = 0.f;
  for (int r = threadIdx.x; r < R; r += blockDim.x) {
    float v = Y[(size_t)r * O + c];
    sum += v; sq += v * v;
  }
  s1[threadIdx.x] = sum; s2[threadIdx.x] = sq;
  __syncthreads();
  for (int k = 128; k > 0; k >>= 1) {
    if (threadIdx.x < k) {
      s1[threadIdx.x] += s1[threadIdx.x + k];
      s2[threadIdx.x] += s2[threadIdx.x + k];
    }
    __syncthreads();
  }
  if (threadIdx.x == 0) {
    float mu = s1[0] / (float)R;
    float var = s2[0] / (float)R - mu * mu;
    if (var < 0.f) var = 0.f;
    stats[2 * c + 0] = mu;
    stats[2 * c + 1] = rsqrtf(var + 1e-5f);
  }
}

__global__ void k_bn_apply(float* __restrict__ Y,
                           const float* __restrict__ stats,
                           const float* __restrict__ g,
                           const float* __restrict__ beta,
                           int R, int O) {
  int idx = blockIdx.x * blockDim.x + threadIdx.x;
  if (idx >= R * O) return;
  int c = idx % O;
  float v = (Y[idx] - stats[2 * c]) * stats[2 * c + 1] * g[c] + beta[c];
  Y[idx] = fmaxf(v, 0.f);
}

// max over nsample: out[b,p,c] = max_s Y[(bp*S+s)*C + c]
__global__ void k_maxpool(const float* __restrict__ Y, int P, int S, int C,
                          float* __restrict__ out) {
  int idx = blockIdx.x * blockDim.x + threadIdx.x;
  if (idx >= BATCH * P * C) return;
  int c  = idx % C;
  int bp = idx / C;
  const float* base = Y + ((size_t)bp * S) * C + c;
  float m = base[0];
  for (int s = 1; s < S; ++s) m = fmaxf(m, base[(size_t)s * C]);
  out[idx] = m;
}

// ---------------------------------------------------------------------------
// 3-NN interpolation: dst[b,i, colOff:colOff+C2] = sum_j w_j * pts2[b,idx_j,:]
// ---------------------------------------------------------------------------
__global__ void k_interp(const float* __restrict__ xyz1, int n1,
                         const float* __restrict__ xyz2, int n2,
                         const float* __restrict__ pts2, int C2,
                         float* __restrict__ dst, int ld, int colOff) {
  int idx = blockIdx.x * blockDim.x + threadIdx.x;
  if (idx >= BATCH * n1) return;
  int b = idx / n1;
  const float* q  = xyz1 + (size_t)idx * 3;
  const float* xb = xyz2 + (size_t)b * n2 * 3;
  float d0 = 1e30f, d1 = 1e30f, d2 = 1e30f;
  int i0 = 0, i1 = 0, i2 = 0;
  for (int j = 0; j < n2; ++j) {
    float dx = xb[j * 3 + 0] - q[0];
    float dy = xb[j * 3 + 1] - q[1];
    float dz = xb[j * 3 + 2] - q[2];
    float d = dx * dx + dy * dy + dz * dz;
    if (d < d0)      { d2 = d1; i2 = i1; d1 = d0; i1 = i0; d0 = d; i0 = j; }
    else if (d < d1) { d2 = d1; i2 = i1; d1 = d; i1 = j; }
    else if (d < d2) { d2 = d;  i2 = j; }
  }
  float w0 = 1.f / (d0 + 1e-8f);
  float w1 = 1.f / (d1 + 1e-8f);
  float w2 = 1.f / (d2 + 1e-8f);
  float ws = w0 + w1 + w2;
  w0 /= ws; w1 /= ws; w2 /= ws;
  const float* p0 = pts2 + ((size_t)b * n2 + i0) * C2;
  const float* p1 = pts2 + ((size_t)b * n2 + i1) * C2;
  const float* p2 = pts2 + ((size_t)b * n2 + i2) * C2;
  float* o = dst + (size_t)idx * ld + colOff;
  for (int c = 0; c < C2; ++c) o[c] = w0 * p0[c] + w1 * p1[c] + w2 * p2[c];
}

// copy skip features into concat columns
__global__ void k_copycols(const float* __restrict__ src, int n1, int C,
                           float* __restrict__ dst, int ld, int colOff) {
  int idx = blockIdx.x * blockDim.x + threadIdx.x;
  if (idx >= BATCH * n1) return;
  const float* s = src + (size_t)idx * C;
  float* o = dst + (size_t)idx * ld + colOff;
  for (int c = 0; c < C; ++c) o[c] = s[c];
}

// log-softmax over 13 classes per row
__global__ void k_logsoftmax(const float* __restrict__ logits,
                             float* __restrict__ out, int R) {
  int r = blockIdx.x * blockDim.x + threadIdx.x;
  if (r >= R) return;
  const float* x = logits + (size_t)r * 13;
  float m = x[0];
#pragma unroll
  for (int c = 1; c < 13; ++c) m = fmaxf(m, x[c]);
  float s = 0.f;
#pragma unroll
  for (int c = 0; c < 13; ++c) s += expf(x[c] - m);
  float ls = logf(s);
  float* o = out + (size_t)r * 13;
#pragma unroll
  for (int c = 0; c < 13; ++c) o[c] = x[c] - m - ls;
}

// l4p (B,16,128) -> out (B,128,16)
__global__ void k_transpose_l4p(const float* __restrict__ l4p,
                                float* __restrict__ out) {
  int idx = blockIdx.x * blockDim.x + threadIdx.x;
  if (idx >= BATCH * 128 * 16) return;
  int b = idx / (128 * 16);
  int rem = idx % (128 * 16);
  int c = rem / 16, p = rem % 16;
  out[idx] = l4p[((size_t)b * 16 + p) * 128 + c];
}

// ---------------------------------------------------------------------------
// host orchestration
// ---------------------------------------------------------------------------
struct Layer { const float *W, *b, *beta, *g; int O, K; };

extern "C" void kernel_launch(void* const* d_in, const int* in_sizes, int n_in,
                              void* d_out, int out_size, void* d_ws, size_t ws_size,
                              hipStream_t stream) {
  (void)in_sizes; (void)out_size; (void)ws_size;

  // --- param leaves (JAX sorted-pytree order: fp, head1, head2, sa, xyz) ---
  auto L = [&](int base, int O, int K) {
    Layer l;
    l.W    = (const float*)d_in[base + 0];
    l.b    = (const float*)d_in[base + 1];
    l.beta = (const float*)d_in[base + 2];
    l.g    = (const float*)d_in[base + 3];
    l.O = O; l.K = K;
    return l;
  };
  Layer fp0[2]  = { L(0,  256, 192), L(4,  256, 256) };
  Layer fp1[2]  = { L(8,  256, 416), L(12, 256, 256) };
  Layer fp2[2]  = { L(16, 256, 656), L(20, 128, 256) };
  Layer fp3[3]  = { L(24, 128, 768), L(28, 128, 128), L(32, 128, 128) };
  Layer head1   =   L(36, 128, 128);
  const float* h2W = (const float*)d_in[40];
  const float* h2b = (const float*)d_in[41];
  Layer sa0[3]  = { L(42, 8, 12),  L(46, 8, 8),   L(50, 16, 8)  };
  Layer sa1[3]  = { L(54, 16, 19), L(58, 16, 16), L(62, 32, 16) };
  Layer sa2[3]  = { L(66, 32, 35), L(70, 32, 32), L(74, 64, 32) };
  Layer sa3[3]  = { L(78, 64, 67), L(82, 64, 64), L(86, 128, 64) };
  const float* xyz_in = (const float*)d_in[n_in - 1];   // 'xyz' sorts last

  // --- workspace layout (float units) ---
  float* ws = (float*)d_ws;
  size_t off = 0;
  float* PTS0 = ws + off; off += (size_t)BATCH * NPTS * 9;
  float* XYZ0 = ws + off; off += (size_t)BATCH * NPTS * 3;
  float* L1X  = ws + off; off += (size_t)BATCH * 1024 * 3;
  float* L1P  = ws + off; off += (size_t)BATCH * 1024 * 128;
  float* L2X  = ws + off; off += (size_t)BATCH * 256 * 3;
  float* L2P  = ws + off; off += (size_t)BATCH * 256 * 256;
  float* L3X  = ws + off; off += (size_t)BATCH * 64 * 3;
  float* L3P  = ws + off; off += (size_t)BATCH * 64 * 256;
  float* L4X  = ws + off; off += (size_t)BATCH * 16 * 3;
  float* L4P  = ws + off; off += (size_t)BATCH * 16 * 128;
  float* STATS = ws + off; off += 1024;
  int* FIDX = (int*)(ws + off); off += (size_t)BATCH * 1024;
  int* GIDX = (int*)(ws + off); off += (size_t)BATCH * 1024 * 32;
  float* BIG0 = ws + off; off += (size_t)BATCH * NPTS * 768;   // 25.2M
  float* BIG1 = ws + off; off += (size_t)BATCH * NPTS * 128;   // 4.2M
  float* BIG2 = ws + off; off += (size_t)BATCH * NPTS * 128;   // 4.2M

  auto conv_bn_relu = [&](const float* X, float* Y, int R, const Layer& Ly) {
    dim3 grid((R + 63) / 64, (Ly.O + 63) / 64);
    k_gemm_bias<<<grid, 128, 0, stream>>>(X, Ly.W, Ly.b, Y, R, Ly.K, Ly.O);
    k_bn_stats<<<Ly.O, 256, 0, stream>>>(Y, R, Ly.O, STATS);
    int tot = R * Ly.O;
    k_bn_apply<<<(tot + 255) / 256, 256, 0, stream>>>(Y, STATS, Ly.g, Ly.beta, R, Ly.O);
  };

  auto run_sa = [&](const float* xyz_src, int Nin, const float* pts_src, int Cpts,
                    int P, float radius, const Layer* Ls, float* lx, float* lp) {
    const int S = 32;
    k_fps<<<BATCH, 256, 0, stream>>>(xyz_src, Nin, P, FIDX);
    int tp = BATCH * P;
    k_gather3<<<(tp + 255) / 256, 256, 0, stream>>>(xyz_src, Nin, FIDX, P, lx);
    k_ballquery<<<(tp + 255) / 256, 256, 0, stream>>>(xyz_src, lx, radius * radius,
                                                      Nin, P, S, GIDX);
    int R = BATCH * P * S;
    k_group<<<(R + 255) / 256, 256, 0, stream>>>(xyz_src, pts_src, Cpts, Nin,
                                                 lx, GIDX, P, S, BIG0);
    conv_bn_relu(BIG0, BIG1, R, Ls[0]);
    conv_bn_relu(BIG1, BIG2, R, Ls[1]);
    conv_bn_relu(BIG2, BIG1, R, Ls[2]);
    int pc = BATCH * P * Ls[2].O;
    k_maxpool<<<(pc + 255) / 256, 256, 0, stream>>>(BIG1, P, S, Ls[2].O, lp);
  };

  auto interp = [&](const float* x1, int n1, const float* x2, int n2,
                    const float* p2, int C2, float* dst, int ld, int colOff) {
    int tot = BATCH * n1;
    k_interp<<<(tot + 255) / 256, 256, 0, stream>>>(x1, n1, x2, n2, p2, C2,
                                                    dst, ld, colOff);
  };
  auto copycols = [&](const float* src, int n1, int C, float* dst, int ld, int colOff) {
    int tot = BATCH * n1;
    k_copycols<<<(tot + 255) / 256, 256, 0, stream>>>(src, n1, C, dst, ld, colOff);
  };

  // ---- input transpose ----
  k_transpose_in<<<(BATCH * NPTS + 255) / 256, 256, 0, stream>>>(xyz_in, PTS0, XYZ0);

  // ---- set-abstraction levels ----
  run_sa(XYZ0, 4096, PTS0, 9,  1024, 0.1f, sa0, L1X, L1P);  // l1p: 16ch
  run_sa(L1X,  1024, L1P, 16, 256,  0.2f, sa1, L2X, L2P);   // l2p: 32ch
  run_sa(L2X,  256,  L2P, 32, 64,   0.4f, sa2, L3X, L3P);   // l3p: 64ch
  run_sa(L3X,  64,   L3P, 64, 16,   0.8f, sa3, L4X, L4P);   // l4p: 128ch

  // ---- FP1: l3 (n1=64), K=192 = interp(l4p,128) + skip l3p(64) ----
  interp(L3X, 64, L4X, 16, L4P, 128, BIG0, 192, 0);
  copycols(L3P, 64, 64, BIG0, 192, 128);
  conv_bn_relu(BIG0, BIG1, BATCH * 64, fp0[0]);
  conv_bn_relu(BIG1, L3P,  BATCH * 64, fp0[1]);             // l3p -> 256ch

  // ---- FP2: l2 (n1=256), K=416 = 128 + 256 + skip 32 ----
  interp(L2X, 256, L4X, 16,  L4P, 128, BIG0, 416, 0);
  interp(L2X, 256, L3X, 64,  L3P, 256, BIG0, 416, 128);
  copycols(L2P, 256, 32, BIG0, 416, 384);
  conv_bn_relu(BIG0, BIG1, BATCH * 256, fp1[0]);
  conv_bn_relu(BIG1, L2P,  BATCH * 256, fp1[1]);            // l2p -> 256ch

  // ---- FP3: l1 (n1=1024), K=656 = 128 + 256 + 256 + skip 16 ----
  interp(L1X, 1024, L4X, 16,  L4P, 128, BIG0, 656, 0);
  interp(L1X, 1024, L3X, 64,  L3P, 256, BIG0, 656, 128);
  interp(L1X, 1024, L2X, 256, L2P, 256, BIG0, 656, 384);
  copycols(L1P, 1024, 16, BIG0, 656, 640);
  conv_bn_relu(BIG0, BIG1, BATCH * 1024, fp2[0]);
  conv_bn_relu(BIG1, L1P,  BATCH * 1024, fp2[1]);           // l1p -> 128ch

  // ---- FP4: l0 (n1=4096), K=768 = 128 + 256 + 256 + 128 ----
  interp(XYZ0, 4096, L4X, 16,   L4P, 128, BIG0, 768, 0);
  interp(XYZ0, 4096, L3X, 64,   L3P, 256, BIG0, 768, 128);
  interp(XYZ0, 4096, L2X, 256,  L2P, 256, BIG0, 768, 384);
  interp(XYZ0, 4096, L1X, 1024, L1P, 128, BIG0, 768, 640);
  int R0 = BATCH * NPTS;
  conv_bn_relu(BIG0, BIG1, R0, fp3[0]);
  conv_bn_relu(BIG1, BIG2, R0, fp3[1]);
  conv_bn_relu(BIG2, BIG1, R0, fp3[2]);

  // ---- head ----
  conv_bn_relu(BIG1, BIG2, R0, head1);
  {
    dim3 grid((R0 + 63) / 64, 1);
    k_gemm_bias<<<grid, 128, 0, stream>>>(BIG2, h2W, h2b, BIG1, R0, 128, 13);
  }
  k_logsoftmax<<<(R0 + 255) / 256, 256, 0, stream>>>(BIG1, (float*)d_out, R0);
  k_transpose_l4p<<<(BATCH * 128 * 16 + 255) / 256, 256, 0, stream>>>(
      L4P, (float*)d_out + (size_t)BATCH * NPTS * 13);
}